// BalanceDiceCoefficientLoss_2731599200955
// MI455X (gfx1250) — compile-verified
//
#include <hip/hip_runtime.h>
#include <math.h>

// ---------------------------------------------------------------------------
// BalanceDiceCoefficientLoss for MI455X (gfx1250, wave32).
// Math reduction: OHEM top-k over negatives == exact selection threshold on
// float bit patterns (monotone for p>=0).  One HBM pass + L2-resident
// refinement passes (inputs ~157MB < 192MB L2).
// ---------------------------------------------------------------------------

#define NBLK 320
#define NTHR 256
#define H1_BINS 8192   // b1 = bits >> 19          (13 bits)
#define H2_BINS 512    // b2 = (bits >> 10) & 0x1FF (9 bits)
#define H3_BINS 1024   // b3 = bits & 0x3FF        (10 bits)

typedef float v2f __attribute__((ext_vector_type(2)));
typedef float v8f __attribute__((ext_vector_type(8)));

// Wave32 sum via V_WMMA_F32_16X16X4_F32: A = ones(16x4) => every row of D is
// the vector of column sums of B; summing the 16 columns (4 xor-shuffles over
// a 16-lane group) yields the total of all 64 B slots.  We place the 32 lane
// values in B vgpr0 and zeros in vgpr1, so the total == wave sum.  This is
// correct for ANY bijective B-slot layout (we only use sum invariance).
__device__ __forceinline__ float wave_sum_wmma(float v) {
  v2f a; a[0] = 1.0f; a[1] = 1.0f;
  v2f b; b[0] = v;    b[1] = 0.0f;
  v8f c = {0.f, 0.f, 0.f, 0.f, 0.f, 0.f, 0.f, 0.f};
  c = __builtin_amdgcn_wmma_f32_16x16x4_f32(false, a, false, b,
                                            (short)0, c, false, false);
  float s = c[0];                 // lane L holds column-sum[L % 16]
  s += __shfl_xor(s, 1, 32);
  s += __shfl_xor(s, 2, 32);
  s += __shfl_xor(s, 4, 32);
  s += __shfl_xor(s, 8, 32);
  return s;                       // every lane: total wave sum
}

// Block (256 thr = 8 waves) sum; result valid on thread 0.
__device__ __forceinline__ float block_sum(float v, float* red8) {
  float w = wave_sum_wmma(v);
  const int lane = threadIdx.x & 31;
  const int wid  = threadIdx.x >> 5;
  __syncthreads();                 // protect red8 reuse across calls
  if (lane == 0) red8[wid] = w;
  __syncthreads();
  float r = 0.f;
  if (threadIdx.x == 0)
    for (int i = 0; i < (int)(blockDim.x >> 5); ++i) r += red8[i];
  return r;
}

// ---------------------------------------------------------------------------
__global__ void k_init(unsigned* __restrict__ w, int n) {
  for (int i = blockIdx.x * blockDim.x + threadIdx.x; i < n;
       i += gridDim.x * blockDim.x)
    w[i] = 0u;
}

// ---------------------------------------------------------------------------
__device__ __forceinline__ void proc1(float p, float t, float mm,
                                      float& posc, float& negc, float& pin,
                                      float& pun, float& asum, unsigned* lh) {
  asum += fabsf(p - t);
  if (mm != 0.0f) {
    if (t != 0.0f) {               // positive (t == 1)
      posc += 1.0f;
      pin  += p;
      pun  += (p + t) + 1e-10f;
    } else {                       // negative
      negc += 1.0f;
      atomicAdd(&lh[__float_as_uint(p) >> 19], 1u);
    }
  }
}

__global__ __launch_bounds__(NTHR) void k_pass1(
    const float* __restrict__ p, const float* __restrict__ t,
    const float* __restrict__ m, int N,
    unsigned* __restrict__ hist1, double* __restrict__ part1) {
  __shared__ unsigned lh[H1_BINS];          // 32 KB LDS histogram
  __shared__ float red8[8];
  for (int i = threadIdx.x; i < H1_BINS; i += NTHR) lh[i] = 0u;
  __syncthreads();

  float posc = 0.f, negc = 0.f, pin = 0.f, pun = 0.f, asum = 0.f;
  const int nvec = N >> 2;
  const float4* p4 = (const float4*)p;
  const float4* t4 = (const float4*)t;
  const float4* m4 = (const float4*)m;
  const int stride = gridDim.x * blockDim.x;
  for (int vi = blockIdx.x * blockDim.x + threadIdx.x; vi < nvec; vi += stride) {
    float4 pv = p4[vi]; float4 tv = t4[vi]; float4 mv = m4[vi];
    proc1(pv.x, tv.x, mv.x, posc, negc, pin, pun, asum, lh);
    proc1(pv.y, tv.y, mv.y, posc, negc, pin, pun, asum, lh);
    proc1(pv.z, tv.z, mv.z, posc, negc, pin, pun, asum, lh);
    proc1(pv.w, tv.w, mv.w, posc, negc, pin, pun, asum, lh);
  }
  if (blockIdx.x == 0 && threadIdx.x == 0)   // scalar tail (N % 4)
    for (int i = (nvec << 2); i < N; ++i)
      proc1(p[i], t[i], m[i], posc, negc, pin, pun, asum, lh);
  __syncthreads();
  for (int i = threadIdx.x; i < H1_BINS; i += NTHR) {
    unsigned c = lh[i];
    if (c) atomicAdd(&hist1[i], c);
  }
  float r0 = block_sum(posc, red8);
  float r1 = block_sum(negc, red8);
  float r2 = block_sum(pin,  red8);
  float r3 = block_sum(pun,  red8);
  float r4 = block_sum(asum, red8);
  if (threadIdx.x == 0) {
    double* o = part1 + (size_t)blockIdx.x * 5;
    o[0] = r0; o[1] = r1; o[2] = r2; o[3] = r3; o[4] = r4;
  }
}

// ---------------------------------------------------------------------------
// Descending scan over a histogram: find bin b* with cum-from-top crossing k.
// Writes b* and k_rem = k - (count strictly above b*) via shared pointers.
__device__ void scan_desc(const unsigned* __restrict__ hist, int nbins,
                          unsigned k, unsigned* sb, unsigned* skr) {
  __shared__ unsigned ssum[NTHR];
  __shared__ unsigned ssuf[NTHR];
  const int chunk = nbins / NTHR;
  const int base  = threadIdx.x * chunk;
  unsigned tot = 0;
  for (int i = 0; i < chunk; ++i) tot += hist[base + i];
  ssum[threadIdx.x] = tot;
  __syncthreads();
  if (threadIdx.x == 0) {
    unsigned c = 0;
    for (int i = NTHR - 1; i >= 0; --i) { ssuf[i] = c; c += ssum[i]; }
  }
  __syncthreads();
  const unsigned se = ssuf[threadIdx.x];
  if (k > se && k <= se + ssum[threadIdx.x]) {   // unique owner chunk
    unsigned c = se;
    for (int b = base + chunk - 1; b >= base; --b) {
      unsigned h = hist[b];
      if (c + h >= k) { *sb = (unsigned)b; *skr = k - c; break; }
      c += h;
    }
  }
}

__global__ __launch_bounds__(NTHR) void k_scan1(
    const double* __restrict__ part1, double* __restrict__ totals,
    unsigned* __restrict__ scal, const unsigned* __restrict__ hist1) {
  __shared__ unsigned sb, skr, kk;
  if (threadIdx.x == 0) {
    double s0 = 0, s1 = 0, s2 = 0, s3 = 0, s4 = 0;
    for (int b = 0; b < NBLK; ++b) {
      const double* o = part1 + (size_t)b * 5;
      s0 += o[0]; s1 += o[1]; s2 += o[2]; s3 += o[3]; s4 += o[4];
    }
    totals[0] = s0; totals[1] = s1; totals[2] = s2; totals[3] = s3; totals[4] = s4;
    // k = (int) min(neg_count_f32, pos_count_f32 * 3.0f)  (exact integers)
    float kf = fminf((float)s1, (float)s0 * 3.0f);
    unsigned k = (unsigned)(int)kf;
    scal[0] = k; kk = k; sb = 0xFFFFFFFFu; skr = 0u;
  }
  __syncthreads();
  if (kk != 0u) scan_desc(hist1, H1_BINS, kk, &sb, &skr);
  __syncthreads();
  if (threadIdx.x == 0) { scal[1] = sb; scal[2] = skr; }
}

// ---------------------------------------------------------------------------
__global__ __launch_bounds__(NTHR) void k_hist2(
    const float* __restrict__ p, const float* __restrict__ t,
    const float* __restrict__ m, int N,
    const unsigned* __restrict__ scal, unsigned* __restrict__ hist2) {
  const unsigned b1 = scal[1];
  if (b1 == 0xFFFFFFFFu) return;             // k == 0
  const int nvec = N >> 2;
  const float4* p4 = (const float4*)p;
  const float4* t4 = (const float4*)t;
  const float4* m4 = (const float4*)m;
  const int stride = gridDim.x * blockDim.x;
  for (int vi = blockIdx.x * blockDim.x + threadIdx.x; vi < nvec; vi += stride) {
    float4 pv = p4[vi]; float4 tv = t4[vi]; float4 mv = m4[vi];
    float pp[4] = {pv.x, pv.y, pv.z, pv.w};
    float tt[4] = {tv.x, tv.y, tv.z, tv.w};
    float mm[4] = {mv.x, mv.y, mv.z, mv.w};
#pragma unroll
    for (int j = 0; j < 4; ++j)
      if (mm[j] != 0.0f && tt[j] == 0.0f) {
        unsigned bits = __float_as_uint(pp[j]);
        if ((bits >> 19) == b1) atomicAdd(&hist2[(bits >> 10) & 0x1FFu], 1u);
      }
  }
  if (blockIdx.x == 0 && threadIdx.x == 0)
    for (int i = (nvec << 2); i < N; ++i)
      if (m[i] != 0.0f && t[i] == 0.0f) {
        unsigned bits = __float_as_uint(p[i]);
        if ((bits >> 19) == b1) atomicAdd(&hist2[(bits >> 10) & 0x1FFu], 1u);
      }
}

__global__ __launch_bounds__(NTHR) void k_scan2(
    const unsigned* __restrict__ hist2, unsigned* __restrict__ scal) {
  __shared__ unsigned sb, skr, kk;
  if (threadIdx.x == 0) { sb = 0xFFFFFFFFu; skr = 0u; kk = scal[2]; }
  __syncthreads();
  if (kk != 0u) scan_desc(hist2, H2_BINS, kk, &sb, &skr);
  __syncthreads();
  if (threadIdx.x == 0) { scal[3] = sb; scal[4] = skr; }
}

__global__ __launch_bounds__(NTHR) void k_hist3(
    const float* __restrict__ p, const float* __restrict__ t,
    const float* __restrict__ m, int N,
    const unsigned* __restrict__ scal, unsigned* __restrict__ hist3) {
  const unsigned b1 = scal[1], b2 = scal[3];
  if (b1 == 0xFFFFFFFFu || b2 == 0xFFFFFFFFu) return;
  const unsigned prefix23 = (b1 << 9) | b2;  // compare against bits >> 10
  const int nvec = N >> 2;
  const float4* p4 = (const float4*)p;
  const float4* t4 = (const float4*)t;
  const float4* m4 = (const float4*)m;
  const int stride = gridDim.x * blockDim.x;
  for (int vi = blockIdx.x * blockDim.x + threadIdx.x; vi < nvec; vi += stride) {
    float4 pv = p4[vi]; float4 tv = t4[vi]; float4 mv = m4[vi];
    float pp[4] = {pv.x, pv.y, pv.z, pv.w};
    float tt[4] = {tv.x, tv.y, tv.z, tv.w};
    float mm[4] = {mv.x, mv.y, mv.z, mv.w};
#pragma unroll
    for (int j = 0; j < 4; ++j)
      if (mm[j] != 0.0f && tt[j] == 0.0f) {
        unsigned bits = __float_as_uint(pp[j]);
        if ((bits >> 10) == prefix23) atomicAdd(&hist3[bits & 0x3FFu], 1u);
      }
  }
  if (blockIdx.x == 0 && threadIdx.x == 0)
    for (int i = (nvec << 2); i < N; ++i)
      if (m[i] != 0.0f && t[i] == 0.0f) {
        unsigned bits = __float_as_uint(p[i]);
        if ((bits >> 10) == prefix23) atomicAdd(&hist3[bits & 0x3FFu], 1u);
      }
}

__global__ __launch_bounds__(NTHR) void k_scan3(
    const unsigned* __restrict__ hist3, unsigned* __restrict__ scal) {
  __shared__ unsigned sb, skr, kk;
  if (threadIdx.x == 0) { sb = 0xFFFFFFFFu; skr = 0u; kk = scal[4]; }
  __syncthreads();
  if (kk != 0u) scan_desc(hist3, H3_BINS, kk, &sb, &skr);
  __syncthreads();
  if (threadIdx.x == 0) {
    unsigned b1 = scal[1], b2 = scal[3], b3 = sb;
    if (b1 == 0xFFFFFFFFu || b2 == 0xFFFFFFFFu || b3 == 0xFFFFFFFFu) {
      scal[5] = 0xFFFFFFFFu; scal[6] = 0u; scal[7] = 0xFFFFFFFFu;
    } else {
      scal[5] = b3; scal[6] = skr;                 // take_eq
      scal[7] = (b1 << 19) | (b2 << 10) | b3;      // exact threshold bits T
    }
  }
}

// ---------------------------------------------------------------------------
__global__ __launch_bounds__(NTHR) void k_sum(
    const float* __restrict__ p, const float* __restrict__ t,
    const float* __restrict__ m, int N,
    const unsigned* __restrict__ scal, double* __restrict__ part2) {
  __shared__ float red8[8];
  const unsigned T = scal[7];                // 0xFFFFFFFF => select nothing
  float acc = 0.f;
  const int nvec = N >> 2;
  const float4* p4 = (const float4*)p;
  const float4* t4 = (const float4*)t;
  const float4* m4 = (const float4*)m;
  const int stride = gridDim.x * blockDim.x;
  for (int vi = blockIdx.x * blockDim.x + threadIdx.x; vi < nvec; vi += stride) {
    float4 pv = p4[vi]; float4 tv = t4[vi]; float4 mv = m4[vi];
    float pp[4] = {pv.x, pv.y, pv.z, pv.w};
    float tt[4] = {tv.x, tv.y, tv.z, tv.w};
    float mm[4] = {mv.x, mv.y, mv.z, mv.w};
#pragma unroll
    for (int j = 0; j < 4; ++j)
      if (mm[j] != 0.0f && tt[j] == 0.0f && __float_as_uint(pp[j]) > T)
        acc += pp[j] + 1e-10f;               // p + t + EPS with t == 0
  }
  if (blockIdx.x == 0 && threadIdx.x == 0)
    for (int i = (nvec << 2); i < N; ++i)
      if (m[i] != 0.0f && t[i] == 0.0f && __float_as_uint(p[i]) > T)
        acc += p[i] + 1e-10f;
  float r = block_sum(acc, red8);
  if (threadIdx.x == 0) part2[blockIdx.x] = (double)r;
}

// ---------------------------------------------------------------------------
__global__ void k_final(const double* __restrict__ totals,
                        const unsigned* __restrict__ scal,
                        const double* __restrict__ part2,
                        float* __restrict__ out, int N) {
  if (threadIdx.x == 0 && blockIdx.x == 0) {
    double sel = 0.0;
    for (int b = 0; b < NBLK; ++b) sel += part2[b];
    unsigned take = scal[6], T = scal[7];
    if (take != 0u)
      sel += (double)take * (double)(__uint_as_float(T) + 1e-10f);
    double posc = totals[0], pin = totals[2], pun = totals[3], asum = totals[4];
    double loss, iou;
    if (posc == 0.0) {
      iou = 0.0;
      loss = asum / (double)N;
    } else {
      iou  = 2.0 * pin / (pun + sel);        // neg_inter == 0 exactly
      loss = 1.0 - iou;
    }
    out[0] = (float)loss;
    out[1] = (float)iou;
  }
}

// ---------------------------------------------------------------------------
// Workspace layout (bytes):
//   0      double totals[8]        (64)
//   64     u32    scal[8]          (32)  [k,b1,k1,b2,k2,b3,take,T]
//   128    u32    hist1[8192]      (32768)
//   32896  u32    hist2[512]       (2048)
//   34944  u32    hist3[1024]      (4096)
//   39040  double part1[NBLK*5]    (12800)
//   51840  double part2[NBLK]      (2560)   total 54400 B
extern "C" void kernel_launch(void* const* d_in, const int* in_sizes, int n_in,
                              void* d_out, int out_size, void* d_ws,
                              size_t ws_size, hipStream_t stream) {
  const float* p = (const float*)d_in[0];
  const float* t = (const float*)d_in[1];
  const float* m = (const float*)d_in[2];
  const int N = in_sizes[0];
  char* ws = (char*)d_ws;
  double*   totals = (double*)(ws + 0);
  unsigned* scal   = (unsigned*)(ws + 64);
  unsigned* hist1  = (unsigned*)(ws + 128);
  unsigned* hist2  = (unsigned*)(ws + 128 + 32768);
  unsigned* hist3  = (unsigned*)(ws + 128 + 32768 + 2048);
  double*   part1  = (double*)(ws + 39040);
  double*   part2  = (double*)(ws + 51840);
  float* out = (float*)d_out;

  k_init <<<40,   NTHR, 0, stream>>>((unsigned*)ws, 39040 / 4);
  k_pass1<<<NBLK, NTHR, 0, stream>>>(p, t, m, N, hist1, part1);
  k_scan1<<<1,    NTHR, 0, stream>>>(part1, totals, scal, hist1);
  k_hist2<<<NBLK, NTHR, 0, stream>>>(p, t, m, N, scal, hist2);
  k_scan2<<<1,    NTHR, 0, stream>>>(hist2, scal);
  k_hist3<<<NBLK, NTHR, 0, stream>>>(p, t, m, N, scal, hist3);
  k_scan3<<<1,    NTHR, 0, stream>>>(hist3, scal);
  k_sum  <<<NBLK, NTHR, 0, stream>>>(p, t, m, N, scal, part2);
  k_final<<<1,    64,   0, stream>>>(totals, scal, part2, out, N);
}